// MultiScaleDeformableAttention_60215441490469
// MI455X (gfx1250) — compile-verified
//
#include <hip/hip_runtime.h>

typedef __attribute__((ext_vector_type(2))) float v2f;
typedef __attribute__((ext_vector_type(8))) float v8f;

// ---- problem constants (static per reference) ----
#define BB   2
#define CCH  256
#define NHH  8
#define DDH  32
#define NLL  4
#define NPP  4
#define NQQ  13294
#define NVV  13294
#define MM   (BB * NQQ)          // 26588 rows for all GEMMs

__device__ __constant__ int c_H[NLL]  = {100, 50, 25, 13};
__device__ __constant__ int c_W[NLL]  = {100, 50, 25, 13};
__device__ __constant__ int c_St[NLL] = {0, 10000, 12500, 13125};

// ============================================================
// GEMM: Out[M,N] = A[M,256] @ W[256,N] + bias[N]
// blockDim=256 (8 waves). Each wave -> one 16x64 output tile.
// W strip [256 x 64] staged in LDS (64KB) via GLOBAL_LOAD_ASYNC_TO_LDS_B128
// (ASYNCcnt DMA). Inner loop: one ds_load_2addr_b32 per WMMA B fragment
// (offset1:64 pairs kk/kk+1 directly into an even VGPR pair) feeding
// V_WMMA_F32_16X16X4_F32.
// ============================================================
__global__ __launch_bounds__(256) void gemm256_bias_wmma(
    const float* __restrict__ A, const float* __restrict__ W,
    const float* __restrict__ bias, float* __restrict__ Out,
    int M, int N)
{
    __shared__ float lds[256 * 64];                 // W[k][n0..n0+63]
    const int tid  = threadIdx.x;
    const int lane = tid & 31;
    const int wave = tid >> 5;
    const int n0   = blockIdx.y * 64;

    // ---- async DMA stage of the W strip: 16B per lane per issue ----
    {
        const unsigned lds_base = (unsigned)(uintptr_t)lds;   // low 32b = LDS offset
        const float* wbase = W + n0;                          // SGPR base (GVS mode)
#pragma unroll
        for (int i = 0; i < 16; ++i) {
            const int f    = tid + i * 256;                   // float4 index 0..4095
            const int krow = f >> 4;                          // 16 float4 per 64-wide row
            const int c4   = f & 15;
            const unsigned goff  = (unsigned)((krow * N + c4 * 4) * 4); // bytes
            const unsigned laddr = lds_base + (unsigned)(f * 16);       // bytes
            asm volatile("global_load_async_to_lds_b128 %0, %1, %2"
                         :: "v"(laddr), "v"(goff), "s"(wbase) : "memory");
        }
        asm volatile("s_wait_asynccnt 0" ::: "memory");
    }
    __syncthreads();

    const int mtile = blockIdx.x * 8 + wave;
    const int m0    = mtile * 16;
    int row_a = m0 + (lane & 15);
    if (row_a >= M) row_a = M - 1;                  // clamp (WMMA needs EXEC all-ones)
    const int koff = (lane >> 4) * 2;               // K sub-pair per lane half
    const int nl   = lane & 15;

    const unsigned lds_base = (unsigned)(uintptr_t)lds;
    v8f acc0 = {}, acc1 = {}, acc2 = {}, acc3 = {};

    for (int k = 0; k < 256; k += 4) {
        // A fragment: lane (m = lane&15) holds A[m][k+koff], A[m][k+koff+1]
        v2f a = *reinterpret_cast<const v2f*>(&A[(size_t)row_a * 256 + k + koff]);
        // B fragments: pair (kk, kk+1) per tile in one dual-address DS load
        const unsigned baddr = lds_base + (unsigned)((((k + koff) * 64) + nl) * 4);
        v2f b0, b1, b2, b3;
        asm volatile(
            "ds_load_2addr_b32 %0, %4 offset1:64\n\t"
            "ds_load_2addr_b32 %1, %4 offset0:16 offset1:80\n\t"
            "ds_load_2addr_b32 %2, %4 offset0:32 offset1:96\n\t"
            "ds_load_2addr_b32 %3, %4 offset0:48 offset1:112\n\t"
            "s_wait_dscnt 0"
            : "=v"(b0), "=v"(b1), "=v"(b2), "=v"(b3)
            : "v"(baddr));
        acc0 = __builtin_amdgcn_wmma_f32_16x16x4_f32(false, a, false, b0, (short)0, acc0, false, false);
        acc1 = __builtin_amdgcn_wmma_f32_16x16x4_f32(false, a, false, b1, (short)0, acc1, false, false);
        acc2 = __builtin_amdgcn_wmma_f32_16x16x4_f32(false, a, false, b2, (short)0, acc2, false, false);
        acc3 = __builtin_amdgcn_wmma_f32_16x16x4_f32(false, a, false, b3, (short)0, acc3, false, false);
    }

    // C/D layout: VGPR j, lanes0-15 -> row m0+j, lanes16-31 -> row m0+8+j
    const int rbase = m0 + 8 * (lane >> 4);
    v8f accs[4] = {acc0, acc1, acc2, acc3};
#pragma unroll
    for (int t = 0; t < 4; ++t) {
        const int col = n0 + t * 16 + nl;
        const float bv = bias[col];
#pragma unroll
        for (int j = 0; j < 8; ++j) {
            int row = rbase + j;
            if (row < M) Out[(size_t)row * N + col] = accs[t][j] + bv;
        }
    }
}

// ============================================================
// Sampling: one wave32 per (b, q, h).
// Every lane mirrors point p = lane&15 (softmax over 16, bilinear corner
// setup), and acts as channel d = lane for the coalesced gathers.
// ============================================================
__global__ __launch_bounds__(256) void msda_sample(
    const float* __restrict__ v,        // [B, Nv, 256] (col = h*32+d)
    const float* __restrict__ off,      // [B*Nq, 256]  (col = h*32 + (l*4+p)*2 + xy)
    const float* __restrict__ logits,   // [B*Nq, 128]  (col = h*16 + l*4+p)
    const float* __restrict__ ref,      // [B, Nq, NL, 2]
    float* __restrict__ out_acc)        // [B*Nq, 256]
{
    const int lane = threadIdx.x & 31;
    const int wid  = (blockIdx.x * blockDim.x + threadIdx.x) >> 5;
    const int total = BB * NQQ * NHH;
    if (wid >= total) return;

    const int b  = wid / (NQQ * NHH);
    const int r  = wid - b * (NQQ * NHH);
    const int q  = r / NHH;
    const int h  = r - q * NHH;
    const int bq = b * NQQ + q;

    // ---- per-point setup ----
    const int pl = lane & 15;
    const int l  = pl >> 2;
    const int Hl = c_H[l], Wl = c_W[l], Sl = c_St[l];

    float logit = logits[(size_t)bq * 128 + h * 16 + pl];
    v2f o = *reinterpret_cast<const v2f*>(&off[(size_t)bq * 256 + h * 32 + pl * 2]);
    const float rx = ref[((size_t)bq * NLL + l) * 2 + 0];
    const float ry = ref[((size_t)bq * NLL + l) * 2 + 1];

    // x = loc_x * W - 0.5 = rx*W + off_x - 0.5 (normalizer cancels)
    const float x = rx * (float)Wl + o[0] - 0.5f;
    const float y = ry * (float)Hl + o[1] - 0.5f;
    const float x0f = floorf(x), y0f = floorf(y);
    const float fx = x - x0f, fy = y - y0f;
    const int x0 = (int)x0f, y0 = (int)y0f;

    float wc[4];
    int   widx[4];
#pragma unroll
    for (int c = 0; c < 4; ++c) {
        const int dx = c & 1, dy = c >> 1;
        const int xi = x0 + dx, yi = y0 + dy;
        const bool valid = (xi >= 0) & (xi < Wl) & (yi >= 0) & (yi < Hl);
        const float wx = dx ? fx : (1.0f - fx);
        const float wy = dy ? fy : (1.0f - fy);
        wc[c] = valid ? (wx * wy) : 0.0f;           // zeros padding mode
        const int xc = min(max(xi, 0), Wl - 1);
        const int yc = min(max(yi, 0), Hl - 1);
        widx[c] = Sl + yc * Wl + xc;
    }

    // ---- softmax over 16 points (butterfly within each half-wave) ----
    float m = logit;
#pragma unroll
    for (int mask = 1; mask < 16; mask <<= 1)
        m = fmaxf(m, __shfl_xor(m, mask, 32));
    float e = __expf(logit - m);
    float s = e;
#pragma unroll
    for (int mask = 1; mask < 16; mask <<= 1)
        s += __shfl_xor(s, mask, 32);
    const float aw = e / s;
#pragma unroll
    for (int c = 0; c < 4; ++c) wc[c] *= aw;        // fold attention weight

    // ---- gather + accumulate: lane = channel d (coalesced 128B reads) ----
    const float* vhead = v + (size_t)b * NVV * 256 + h * 32;
    float acc = 0.0f;
    for (int p = 0; p < 16; ++p) {
#pragma unroll
        for (int c = 0; c < 4; ++c) {
            const float w = __shfl(wc[c], p, 32);
            const int   id = __shfl(widx[c], p, 32);
            if (w != 0.0f)
                acc += w * vhead[(size_t)id * 256 + lane];
        }
    }
    out_acc[(size_t)bq * 256 + h * 32 + lane] = acc;
}

// ============================================================
// launcher
// ============================================================
extern "C" void kernel_launch(void* const* d_in, const int* in_sizes, int n_in,
                              void* d_out, int out_size, void* d_ws, size_t ws_size,
                              hipStream_t stream) {
    const float* query  = (const float*)d_in[0];
    const float* value  = (const float*)d_in[1];
    const float* refpts = (const float*)d_in[2];
    // d_in[3] spatial_shapes, d_in[4] level_start_index: compile-time constants
    const float* W_off  = (const float*)d_in[5];
    const float* b_off  = (const float*)d_in[6];
    const float* W_attn = (const float*)d_in[7];
    const float* b_attn = (const float*)d_in[8];
    const float* W_val  = (const float*)d_in[9];
    const float* b_val  = (const float*)d_in[10];
    const float* W_out  = (const float*)d_in[11];
    const float* b_out  = (const float*)d_in[12];
    float* out = (float*)d_out;

    // workspace partition (floats): v | off | attn | acc  (~95 MB)
    float* ws       = (float*)d_ws;
    float* v_buf    = ws;
    float* off_buf  = v_buf   + (size_t)MM * 256;
    float* attn_buf = off_buf + (size_t)MM * 256;
    float* acc_buf  = attn_buf + (size_t)MM * 128;

    const int M = MM;                               // 26588
    const int mtiles = (M + 15) / 16;               // 1662
    const int gx = (mtiles + 7) / 8;                // 208 blocks (8 waves each)

    // 1) value projection  v = value @ W_val + b_val
    gemm256_bias_wmma<<<dim3(gx, 4), 256, 0, stream>>>(value, W_val, b_val, v_buf, M, 256);
    // 2) sampling offsets  off = query @ W_off + b_off
    gemm256_bias_wmma<<<dim3(gx, 4), 256, 0, stream>>>(query, W_off, b_off, off_buf, M, 256);
    // 3) attention logits  attn = query @ W_attn + b_attn
    gemm256_bias_wmma<<<dim3(gx, 2), 256, 0, stream>>>(query, W_attn, b_attn, attn_buf, M, 128);
    // 4) softmax + bilinear sampling + head accumulation
    {
        const int waves = BB * NQQ * NHH;           // 212704
        const int blocks = (waves + 7) / 8;
        msda_sample<<<blocks, 256, 0, stream>>>(v_buf, off_buf, attn_buf, refpts, acc_buf);
    }
    // 5) output projection
    gemm256_bias_wmma<<<dim3(gx, 4), 256, 0, stream>>>(acc_buf, W_out, b_out, out, M, 256);
}